// AttentionGraphUnlearning_44057774522830
// MI455X (gfx1250) — compile-verified
//
#include <hip/hip_runtime.h>
#include <math.h>

// N=100000 nodes, D=64, H=2 heads x HD=32; E_drp=1M, E_pk=2M.
// SpMM passes dominate (~8.4GB traffic -> ~360us @ 23.3TB/s, bandwidth bound).
// Dense GEMMs (qkv 100000x64x192, out/mlp 100000x64x64) use fp32 WMMA
// V_WMMA_F32_16X16X4_F32 to preserve reference precision.

typedef __attribute__((ext_vector_type(2))) float v2f;
typedef __attribute__((ext_vector_type(8))) float v8f;

#define DDIM 64

__device__ __forceinline__ void atomAddF(float* p, float v) {
  __hip_atomic_fetch_add(p, v, __ATOMIC_RELAXED, __HIP_MEMORY_SCOPE_AGENT);
}

__device__ __forceinline__ float lrelu_f(float x, float s) {
  return x >= 0.f ? x : s * x;
}

// ---------------- SpMM: y[row[e],:] += w_e * x[col[e],:] ----------------
// One wave per edge; lane handles dims {lane, lane+32}. Edge weight is
// wave-uniform so the zero-skip keeps EXEC uniform.
__global__ void spmm_edge_kernel(const int* __restrict__ row,
                                 const int* __restrict__ col,
                                 const float* __restrict__ val,
                                 const unsigned char* __restrict__ mask, // nullable
                                 float scale,
                                 const float* __restrict__ x,
                                 float* __restrict__ y,
                                 int E) {
  int wid = (int)((blockIdx.x * blockDim.x + threadIdx.x) >> 5);
  if (wid >= E) return;
  int lane = threadIdx.x & 31;
  float w = val[wid] * scale;
  if (mask && !mask[wid]) w = 0.f;
  if (w == 0.f) return;  // wave-uniform
  int r = row[wid], c = col[wid];
  const float* xs = x + (size_t)c * DDIM;
  float* yd = y + (size_t)r * DDIM;
  atomAddF(yd + lane,      w * xs[lane]);
  atomAddF(yd + lane + 32, w * xs[lane + 32]);
}

// ---------------- fp32 WMMA GEMM: C[M,Nc] = A[M,K] @ B[K,Nc] ----------------
// One wave per 16x16 tile, K consumed 4 at a time via V_WMMA_F32_16X16X4_F32.
// A 16x4 layout: lanes 0-15 hold M=lane with K pair (k0,k0+1), lanes 16-31
// hold K pair (k0+2,k0+3). B 4x16 mirrors with N=lane&15. D: VGPR v -> row
// (half*8+v), col lane&15.
__global__ void wmma_gemm_f32_kernel(const float* __restrict__ A,
                                     const float* __restrict__ B,
                                     float* __restrict__ C,
                                     int M, int K, int Nc) {
  int wavesPerBlock = blockDim.x >> 5;
  int tile = blockIdx.x * wavesPerBlock + (int)(threadIdx.x >> 5);
  int tilesN = Nc >> 4;
  int totalTiles = (M >> 4) * tilesN;
  if (tile >= totalTiles) return;  // wave-uniform: EXEC stays all-ones
  int tm = (tile / tilesN) << 4;
  int tn = (tile % tilesN) << 4;
  int lane = threadIdx.x & 31;
  int mn   = lane & 15;
  int half = lane >> 4;
  v8f acc = {};
  const float* arow = A + (size_t)(tm + mn) * K;
  for (int k0 = 0; k0 < K; k0 += 4) {
    int ka = k0 + 2 * half;
    v2f a, b;
    a.x = arow[ka];
    a.y = arow[ka + 1];
    b.x = B[(size_t)ka * Nc + tn + mn];
    b.y = B[(size_t)(ka + 1) * Nc + tn + mn];
    acc = __builtin_amdgcn_wmma_f32_16x16x4_f32(
        /*neg_a=*/false, a, /*neg_b=*/false, b,
        /*c_mod=*/(short)0, acc, /*reuse_a=*/false, /*reuse_b=*/false);
  }
#pragma unroll
  for (int v = 0; v < 8; ++v) {
    C[(size_t)(tm + half * 8 + v) * Nc + tn + mn] = acc[v];
  }
}

// ---------------- attention scores (per edge) ----------------
// qkv row stride 192: Q cols [0,64), K cols [64,128), V cols [128,192).
__global__ void attn_score_kernel(const int* __restrict__ row,
                                  const int* __restrict__ col,
                                  const float* __restrict__ qkv,
                                  float* __restrict__ exps,
                                  float* __restrict__ denom,
                                  int E) {
  int wid = (int)((blockIdx.x * blockDim.x + threadIdx.x) >> 5);
  if (wid >= E) return;
  int lane = threadIdx.x & 31;
  int r = row[wid], c = col[wid];
  const float* q = qkv + (size_t)r * 192;
  const float* k = qkv + (size_t)c * 192 + 64;
  float p0 = q[lane] * k[lane];
  float p1 = q[lane + 32] * k[lane + 32];
  for (int o = 16; o > 0; o >>= 1) {
    p0 += __shfl_xor(p0, o, 32);
    p1 += __shfl_xor(p1, o, 32);
  }
  if (lane == 0) {
    const float inv = 0.17677669529663687f;  // 1/sqrt(32)
    float s0 = lrelu_f(p0 * inv, 0.2f);
    float s1 = lrelu_f(p1 * inv, 0.2f);
    s0 = fminf(fmaxf(s0, -20.f), 20.f);
    s1 = fminf(fmaxf(s1, -20.f), 20.f);
    float e0 = expf(s0);
    float e1 = expf(s1);
    exps[2 * (size_t)wid]     = e0;
    exps[2 * (size_t)wid + 1] = e1;
    atomAddF(denom + 2 * (size_t)r,     e0);
    atomAddF(denom + 2 * (size_t)r + 1, e1);
  }
}

// ---------------- attention aggregation (per edge) ----------------
__global__ void attn_agg_kernel(const int* __restrict__ row,
                                const int* __restrict__ col,
                                const float* __restrict__ qkv,
                                const float* __restrict__ exps,
                                const float* __restrict__ denom,
                                float* __restrict__ agg,
                                int E) {
  int wid = (int)((blockIdx.x * blockDim.x + threadIdx.x) >> 5);
  if (wid >= E) return;
  int lane = threadIdx.x & 31;
  int r = row[wid], c = col[wid];
  float w0 = exps[2 * (size_t)wid]     / (denom[2 * (size_t)r]     + 1e-10f);
  float w1 = exps[2 * (size_t)wid + 1] / (denom[2 * (size_t)r + 1] + 1e-10f);
  const float* v = qkv + (size_t)c * 192 + 128;
  float* ar = agg + (size_t)r * DDIM;
  atomAddF(ar + lane,      w0 * v[lane]);
  atomAddF(ar + lane + 32, w1 * v[lane + 32]);
}

// ---------------- LayerNorm (one wave per row) ----------------
__global__ void layernorm_kernel(const float* __restrict__ z,
                                 const float* __restrict__ gamma,
                                 const float* __restrict__ beta,
                                 float* __restrict__ out, int Nn) {
  int wid = (int)((blockIdx.x * blockDim.x + threadIdx.x) >> 5);
  if (wid >= Nn) return;
  int lane = threadIdx.x & 31;
  const float* zr = z + (size_t)wid * DDIM;
  float z0 = zr[lane], z1 = zr[lane + 32];
  float s = z0 + z1;
  for (int o = 16; o > 0; o >>= 1) s += __shfl_xor(s, o, 32);
  float mu = s * (1.f / 64.f);
  float d0 = z0 - mu, d1 = z1 - mu;
  float sq = d0 * d0 + d1 * d1;
  for (int o = 16; o > 0; o >>= 1) sq += __shfl_xor(sq, o, 32);
  float rstd = rsqrtf(sq * (1.f / 64.f) + 1e-5f);
  float* orow = out + (size_t)wid * DDIM;
  orow[lane]      = gamma[lane]      * d0 * rstd + beta[lane];
  orow[lane + 32] = gamma[lane + 32] * d1 * rstd + beta[lane + 32];
}

// ---------------- elementwise helpers ----------------
__global__ void add3_kernel(const float* __restrict__ a, const float* __restrict__ b,
                            const float* __restrict__ c, float* __restrict__ o, size_t n) {
  size_t i = (size_t)blockIdx.x * blockDim.x + threadIdx.x;
  if (i < n) o[i] = a[i] + b[i] + c[i];
}

__global__ void axpy_kernel(float alpha, const float* __restrict__ x,
                            float* __restrict__ y, size_t n) {
  size_t i = (size_t)blockIdx.x * blockDim.x + threadIdx.x;
  if (i < n) y[i] += alpha * x[i];
}

__global__ void wd_init_kernel(const float* __restrict__ fnl, const float* __restrict__ wr,
                               float* __restrict__ o, size_t n) {
  size_t i = (size_t)blockIdx.x * blockDim.x + threadIdx.x;
  if (i < n) o[i] = fnl[i] * wr[i >> 6];
}

__global__ void bias_lrelu_kernel(float* __restrict__ x, const float* __restrict__ b,
                                  float slope, size_t n) {
  size_t i = (size_t)blockIdx.x * blockDim.x + threadIdx.x;
  if (i < n) x[i] = lrelu_f(x[i] + b[i & 63], slope);
}

__global__ void final_out_kernel(const float* __restrict__ ini, const float* __restrict__ x,
                                 const float* __restrict__ b, float slope,
                                 float* __restrict__ o, size_t n) {
  size_t i = (size_t)blockIdx.x * blockDim.x + threadIdx.x;
  if (i < n) o[i] = ini[i] + lrelu_f(x[i] + b[i & 63], slope);
}

// ---------------- orchestration ----------------
extern "C" void kernel_launch(void* const* d_in, const int* in_sizes, int n_in,
                              void* d_out, int out_size, void* d_ws, size_t ws_size,
                              hipStream_t stream) {
  const float* edge_embeds = (const float*)d_in[0];
  const float* ini         = (const float*)d_in[1];
  const float* fnl         = (const float*)d_in[2];
  const float* wr          = (const float*)d_in[3];
  const float* W_qkv       = (const float*)d_in[4];
  const float* W_out       = (const float*)d_in[5];
  const float* gamma       = (const float*)d_in[6];
  const float* beta        = (const float*)d_in[7];
  const float* W_mlp       = (const float*)d_in[8];   // [2,64,64]
  const float* b_mlp       = (const float*)d_in[9];   // [2,64]
  const float* val_drp     = (const float*)d_in[10];
  const float* val_pk      = (const float*)d_in[11];
  const int*   row_drp     = (const int*)d_in[12];
  const int*   col_drp     = (const int*)d_in[13];
  const int*   row_pk      = (const int*)d_in[14];
  const int*   col_pk      = (const int*)d_in[15];
  const unsigned char* drop_mask = (const unsigned char*)d_in[16]; // [2,E1] bool

  const int Nn = in_sizes[3];                 // withdraw_rate: N elements
  const int E1 = in_sizes[10];                // drp edges
  const int E2 = in_sizes[11];                // pk edges
  const size_t ND = (size_t)Nn * DDIM;

  float* out   = (float*)d_out;
  float* tuned = out;
  float* gnn0  = out + ND;
  float* gnn1  = out + 2 * ND;
  float* hyp0  = out + 3 * ND;
  float* hyp1  = out + 4 * ND;

  float* ws    = (float*)d_ws;
  float* Hold  = ws;                  // [N,64]
  float* bufA  = ws + ND;             // [N,64]
  float* bufB  = ws + 2 * ND;         // [N,64]
  float* qkv   = ws + 3 * ND;         // [N,192]
  float* exps  = ws + 6 * ND;         // [E1,2]
  float* denom = exps + 2 * (size_t)E1;  // [N,2]

  const unsigned TPB = 256;
  auto eGrid = [&](int E) { return dim3((unsigned)(((size_t)E * 32 + TPB - 1) / TPB)); };
  auto nGrid = [&](size_t n) { return dim3((unsigned)((n + TPB - 1) / TPB)); };
  dim3 rowGrid((unsigned)(((size_t)Nn * 32 + TPB - 1) / TPB));
  const float KEEP = 0.95f, OWR = 0.1f, LEAKY = 0.5f;

#define MEMZ(p, nfloats) hipMemsetAsync((void*)(p), 0, (size_t)(nfloats) * sizeof(float), stream)
#define SPMM(r, c, v, m, sc, xin, yout, E) \
  spmm_edge_kernel<<<eGrid(E), TPB, 0, stream>>>(r, c, v, m, sc, xin, yout, E)

  // --- GNN layers over drp adjacency (tem + hyper per layer) ---
  MEMZ(gnn0, ND); SPMM(row_drp, col_drp, val_drp, (const unsigned char*)nullptr, 1.0f, edge_embeds, gnn0, E1);
  MEMZ(hyp0, ND); SPMM(row_drp, col_drp, val_drp, drop_mask, 1.0f / KEEP, edge_embeds, hyp0, E1);
  MEMZ(gnn1, ND); SPMM(row_drp, col_drp, val_drp, (const unsigned char*)nullptr, 1.0f, gnn0, gnn1, E1);
  MEMZ(hyp1, ND); SPMM(row_drp, col_drp, val_drp, drop_mask + (size_t)E1, 1.0f / KEEP, gnn0, hyp1, E1);

  // --- lat_sum = edge_embeds + tem0 + tem1 ---
  add3_kernel<<<nGrid(ND), TPB, 0, stream>>>(edge_embeds, gnn0, gnn1, bufA, ND);

  // --- edge_out = pk^2 @ lat_sum -> accumulate directly into H_old ---
  MEMZ(bufB, ND); SPMM(row_pk, col_pk, val_pk, (const unsigned char*)nullptr, 1.0f, bufA, bufB, E2);
  MEMZ(Hold, ND); SPMM(row_pk, col_pk, val_pk, (const unsigned char*)nullptr, 1.0f, bufB, Hold, E2);

  // --- wd = drp^2 @ (fnl * withdraw); H_old += -OWR * wd ---
  wd_init_kernel<<<nGrid(ND), TPB, 0, stream>>>(fnl, wr, bufA, ND);
  MEMZ(bufB, ND); SPMM(row_drp, col_drp, val_drp, (const unsigned char*)nullptr, 1.0f, bufA, bufB, E1);
  MEMZ(bufA, ND); SPMM(row_drp, col_drp, val_drp, (const unsigned char*)nullptr, 1.0f, bufB, bufA, E1);
  axpy_kernel<<<nGrid(ND), TPB, 0, stream>>>(-OWR, bufA, Hold, ND);

  // --- h_bar = drp^2 @ ini; H_old += h_bar ---
  MEMZ(bufB, ND); SPMM(row_drp, col_drp, val_drp, (const unsigned char*)nullptr, 1.0f, ini, bufB, E1);
  MEMZ(bufA, ND); SPMM(row_drp, col_drp, val_drp, (const unsigned char*)nullptr, 1.0f, bufB, bufA, E1);
  axpy_kernel<<<nGrid(ND), TPB, 0, stream>>>(1.0f, bufA, Hold, ND);

  // --- e_bar = pk^2 @ ini; H_old -= e_bar ---
  MEMZ(bufB, ND); SPMM(row_pk, col_pk, val_pk, (const unsigned char*)nullptr, 1.0f, ini, bufB, E2);
  MEMZ(bufA, ND); SPMM(row_pk, col_pk, val_pk, (const unsigned char*)nullptr, 1.0f, bufB, bufA, E2);
  axpy_kernel<<<nGrid(ND), TPB, 0, stream>>>(-1.0f, bufA, Hold, ND);

  // --- qkv = H_old @ W_qkv  (WMMA fp32) ---
  {
    int tiles = (Nn / 16) * (192 / 16);
    wmma_gemm_f32_kernel<<<dim3((unsigned)((tiles + 7) / 8)), TPB, 0, stream>>>(
        Hold, W_qkv, qkv, Nn, 64, 192);
  }

  // --- attention: scores + softmax denominators, then weighted aggregation ---
  MEMZ(denom, 2 * (size_t)Nn);
  attn_score_kernel<<<eGrid(E1), TPB, 0, stream>>>(row_drp, col_drp, qkv, exps, denom, E1);
  MEMZ(bufA, ND);  // agg
  attn_agg_kernel<<<eGrid(E1), TPB, 0, stream>>>(row_drp, col_drp, qkv, exps, denom, bufA, E1);

  // --- H_attn = agg @ W_out (WMMA); z = H_old + H_attn ---
  {
    int tiles = (Nn / 16) * (64 / 16);
    wmma_gemm_f32_kernel<<<dim3((unsigned)((tiles + 7) / 8)), TPB, 0, stream>>>(
        bufA, W_out, bufB, Nn, 64, 64);
  }
  axpy_kernel<<<nGrid(ND), TPB, 0, stream>>>(1.0f, bufB, Hold, ND);

  // --- LayerNorm -> delta ---
  layernorm_kernel<<<rowGrid, TPB, 0, stream>>>(Hold, gamma, beta, bufA, Nn);

  // --- MLP layer 0: bufB = lrelu(bufA @ W_mlp0 + b0) ---
  {
    int tiles = (Nn / 16) * (64 / 16);
    wmma_gemm_f32_kernel<<<dim3((unsigned)((tiles + 7) / 8)), TPB, 0, stream>>>(
        bufA, W_mlp, bufB, Nn, 64, 64);
  }
  bias_lrelu_kernel<<<nGrid(ND), TPB, 0, stream>>>(bufB, b_mlp, LEAKY, ND);

  // --- MLP layer 1 + residual: tuned = ini + lrelu(bufB @ W_mlp1 + b1) ---
  {
    int tiles = (Nn / 16) * (64 / 16);
    wmma_gemm_f32_kernel<<<dim3((unsigned)((tiles + 7) / 8)), TPB, 0, stream>>>(
        bufB, W_mlp + 64 * 64, bufA, Nn, 64, 64);
  }
  final_out_kernel<<<nGrid(ND), TPB, 0, stream>>>(ini, bufA, b_mlp + 64, LEAKY, tuned, ND);

#undef MEMZ
#undef SPMM
}